// HistogramLayer_1511828488271
// MI455X (gfx1250) — compile-verified
//
#include <hip/hip_runtime.h>
#include <stdint.h>
#include <math.h>

// ---------------------------------------------------------------------------
// Per-image 256-bin histogram of luma(gray) with normalization.
// Input : x [64, 512, 512, 3] float32  (201.3 MB -> bandwidth bound, ~8.6us floor)
// Output: [64, 256] float32
//
// CDNA5 paths: GLOBAL_LOAD_ASYNC_TO_LDS_B128 (async data mover, ASYNCcnt,
// non-temporal hint since the stream is single-pass), double-buffered LDS
// tiles, per-wave privatized DS-atomic histograms. No WMMA: K=3 dot at
// ~0.5 flop/byte is >10x below any matrix-op ceiling; this is pure bandwidth.
// ---------------------------------------------------------------------------

#define NBINS          256
#define THREADS        256
#define NWAVES         (THREADS / 32)
#define HSTRIDE        260                  // padded replica stride (dwords)
#define TILE_PIX       2048
#define TILE_FLOATS    (TILE_PIX * 3)       // 6144 floats = 24 KB
#define COPIES_PER_TILE (TILE_FLOATS / (THREADS * 4))   // 6 x b128 per thread
#define H_IMG          512
#define W_IMG          512
#define C_IMG          3
#define PIX_PER_IMG    (H_IMG * W_IMG)              // 262144
#define BLOCKS_PER_IMG 32
#define PIX_PER_BLOCK  (PIX_PER_IMG / BLOCKS_PER_IMG)   // 8192
#define TILES_PER_BLK  (PIX_PER_BLOCK / TILE_PIX)       // 4

// ---- CDNA5 async global -> LDS copy (16 bytes per lane), non-temporal -----
__device__ __forceinline__ void async_copy_b128_nt(uint32_t lds_byte_addr,
                                                   const float* gptr) {
    asm volatile("global_load_async_to_lds_b128 %0, %1, off th:TH_LOAD_NT"
                 :: "v"(lds_byte_addr), "v"(gptr)
                 : "memory");
}

__device__ __forceinline__ void wait_asynccnt_le(int) = delete;

__device__ __forceinline__ void wait_asynccnt_le6() {
#if __has_builtin(__builtin_amdgcn_s_wait_asynccnt)
    __builtin_amdgcn_s_wait_asynccnt(COPIES_PER_TILE);
#else
    asm volatile("s_wait_asynccnt 0x6" ::: "memory");
#endif
}

__device__ __forceinline__ void wait_asynccnt_0() {
#if __has_builtin(__builtin_amdgcn_s_wait_asynccnt)
    __builtin_amdgcn_s_wait_asynccnt(0);
#else
    asm volatile("s_wait_asynccnt 0x0" ::: "memory");
#endif
}

// Flat LDS address -> 32-bit LDS byte offset (low 32 bits of flat aperture addr)
__device__ __forceinline__ uint32_t lds_off(const void* p) {
    return (uint32_t)(uintptr_t)p;
}

// ---------------------------------------------------------------------------
// Main streaming histogram kernel (placed first so its asm shows in snippet).
__global__ __launch_bounds__(THREADS)
void hist_kernel(const float* __restrict__ x, uint32_t* __restrict__ ghist) {
    __shared__ float    stage[2][TILE_FLOATS];      // 48 KB double buffer
    __shared__ uint32_t hsub[NWAVES * HSTRIDE];     // 8 per-wave replicas

    const int t     = threadIdx.x;
    const int wave  = t >> 5;
    const int img   = blockIdx.x / BLOCKS_PER_IMG;
    const int blk   = blockIdx.x % BLOCKS_PER_IMG;

    for (int i = t; i < NWAVES * HSTRIDE; i += THREADS) hsub[i] = 0u;

    const float* base = x + (size_t)img * (PIX_PER_IMG * C_IMG)
                          + (size_t)blk * (PIX_PER_BLOCK * C_IMG);

    const uint32_t lds0   = lds_off(&stage[0][0]);
    const uint32_t lds1   = lds_off(&stage[1][0]);
    const uint32_t lane16 = (uint32_t)t * 16u;

    // Prologue: async-prefetch tile 0 (6 x 4KB coalesced b128 copies)
    {
        const float* g = base + t * 4;
        #pragma unroll
        for (int k = 0; k < COPIES_PER_TILE; ++k)
            async_copy_b128_nt(lds0 + lane16 + 4096u * k, g + 1024 * k);
    }

    uint32_t* myhist = &hsub[wave * HSTRIDE];

    for (int tile = 0; tile < TILES_PER_BLK; ++tile) {
        const float* scur = (tile & 1) ? &stage[1][0] : &stage[0][0];
        if (tile + 1 < TILES_PER_BLK) {
            // prefetch next tile into the other buffer while this one is used
            const uint32_t nxt = (tile & 1) ? lds0 : lds1;
            const float* g = base + (size_t)(tile + 1) * TILE_FLOATS + t * 4;
            #pragma unroll
            for (int k = 0; k < COPIES_PER_TILE; ++k)
                async_copy_b128_nt(nxt + lane16 + 4096u * k, g + 1024 * k);
            wait_asynccnt_le6();   // current tile's copies retired (in-order)
        } else {
            wait_asynccnt_0();
        }
        __syncthreads();           // all waves' copies for this tile visible

        #pragma unroll
        for (int p = 0; p < TILE_PIX / THREADS; ++p) {
            const int pix = p * THREADS + t;   // stride keeps b32 LDS reads conflict-free
            const float r = scur[pix * 3 + 0];
            const float g = scur[pix * 3 + 1];
            const float b = scur[pix * 3 + 2];
            const float gray = fmaf(0.2989f, r, fmaf(0.5870f, g, 0.1140f * b));
            // idx = clip(floor(gray / (255/256)), 0, 255)
            int idx = (int)floorf(gray * (256.0f / 255.0f));
            idx = idx < 0 ? 0 : (idx > (NBINS - 1) ? (NBINS - 1) : idx);
            atomicAdd(&myhist[idx], 1u);       // ds_add_u32 (per-wave replica)
        }
        __syncthreads();           // everyone done reading before buffer reuse
    }

    __syncthreads();
    // bin t: sum the 8 wave replicas, one global atomic per bin per block
    uint32_t v = 0u;
    #pragma unroll
    for (int w = 0; w < NWAVES; ++w) v += hsub[w * HSTRIDE + t];
    atomicAdd(&ghist[img * NBINS + t], v);
}

// ---------------------------------------------------------------------------
__global__ void zero_hist_kernel(uint32_t* __restrict__ g, int n) {
    int i = blockIdx.x * blockDim.x + threadIdx.x;
    if (i < n) g[i] = 0u;
}

__global__ __launch_bounds__(NBINS)
void norm_kernel(const uint32_t* __restrict__ ghist, float* __restrict__ out) {
    __shared__ float ssum[NBINS];
    const int b = blockIdx.x, t = threadIdx.x;
    const float v = (float)ghist[b * NBINS + t];
    ssum[t] = v;
    __syncthreads();
    for (int s = NBINS / 2; s > 0; s >>= 1) {
        if (t < s) ssum[t] += ssum[t + s];
        __syncthreads();
    }
    const float s = ssum[0];
    out[b * NBINS + t] = (s > 0.0f) ? v / fmaxf(s, 1.0f) : v;
}

// ---------------------------------------------------------------------------
extern "C" void kernel_launch(void* const* d_in, const int* in_sizes, int n_in,
                              void* d_out, int out_size, void* d_ws, size_t ws_size,
                              hipStream_t stream) {
    const float* x   = (const float*)d_in[0];
    float*       out = (float*)d_out;
    uint32_t*    gh  = (uint32_t*)d_ws;

    const int n = in_sizes[0];                       // B*H*W*C
    const int B = n / (H_IMG * W_IMG * C_IMG);       // 64
    const int nh = B * NBINS;

    zero_hist_kernel<<<(nh + 255) / 256, 256, 0, stream>>>(gh, nh);
    hist_kernel<<<B * BLOCKS_PER_IMG, THREADS, 0, stream>>>(x, gh);
    norm_kernel<<<B, NBINS, 0, stream>>>(gh, out);
}